// QuadraticAttentionModule_58548994179563
// MI455X (gfx1250) — compile-verified
//
#include <hip/hip_runtime.h>

// Problem constants (from reference)
#define IN_DIM   1024
#define N_QUAD   524800
#define K_TOT    (IN_DIM + N_QUAD)      // 525824 = 2054 * 256 (exact)
#define OUT_DIM  512
#define BATCH    32

// Tiling
#define THREADS         512
#define WAVES           16
#define OT              32               // output columns per workgroup
#define KT              256              // staged K per LDS tile (divides K_TOT)
#define TILES_PER_CHUNK 32
#define KC              (KT * TILES_PER_CHUNK)       // 8192 K per workgroup
#define NKC             ((K_TOT + KC - 1) / KC)      // 65 K-chunks (last has 6 tiles)
#define XS_STRIDE       1025             // pad => conflict-free x reads
#define WSTRIDE         260              // pad (260%64==4) => conflict-free W frag reads
#define QSTEP           64               // per-lane k stride (WAVES*4)

typedef float v2f __attribute__((ext_vector_type(2)));
typedef float v8f __attribute__((ext_vector_type(8)));

__device__ __forceinline__ int triu_row_start(int r) {
    return r * IN_DIM - ((r * (r - 1)) >> 1);     // r*N - r*(r-1)/2
}

// one-time solve: (r,c) for triu flat index q (row-major incl. diagonal)
__device__ __forceinline__ void triu_solve(int q, int& r, int& c) {
    float disc = 2049.0f * 2049.0f - 8.0f * (float)q;   // exact: < 2^23
    int rr = (int)((2049.0f - sqrtf(disc)) * 0.5f);
    rr = max(0, min(rr, IN_DIM - 1));
    if (q < triu_row_start(rr)) --rr;
    else if (rr + 1 < IN_DIM && q >= triu_row_start(rr + 1)) ++rr;
    if (q < triu_row_start(rr)) --rr;
    else if (rr + 1 < IN_DIM && q >= triu_row_start(rr + 1)) ++rr;
    r = rr;
    c = rr + (q - triu_row_start(rr));
}

__global__ void bias_init_kernel(const float* __restrict__ bias,
                                 float* __restrict__ out) {
    int i = blockIdx.x * blockDim.x + threadIdx.x;
    if (i < BATCH * OUT_DIM) out[i] = bias[i & (OUT_DIM - 1)];
}

__global__ __launch_bounds__(THREADS)
void quad_gemm_kernel(const float* __restrict__ X,   // [32,1024]
                      const float* __restrict__ W,   // [512, K_TOT] row-major
                      float* __restrict__ out)       // [32,512]
{
    __shared__ float xs[BATCH * XS_STRIDE];          // 131200 B
    __shared__ float wtile[OT * WSTRIDE];            // 33280 B, [row][kt] padded
    __shared__ float accs[BATCH * OT];               // 4096 B

    const int tid  = threadIdx.x;
    const int lane = tid & 31;
    const int wave = tid >> 5;                       // 0..15
    const int o0   = blockIdx.x * OT;                // 32 output columns (W rows)
    const int kc0  = (int)blockIdx.y * KC;           // K-chunk base
    const int ntiles = min(TILES_PER_CHUNK, (K_TOT - kc0) / KT);  // K_TOT%KT==0

    // ---- stage x into LDS (padded rows), zero LDS accumulators ----
    for (int i = tid; i < BATCH * (IN_DIM / 4); i += THREADS) {
        int b  = i / (IN_DIM / 4);
        int c4 = i % (IN_DIM / 4);
        const float4 v = ((const float4*)X)[(size_t)b * (IN_DIM / 4) + c4];
        float* dst = &xs[b * XS_STRIDE + c4 * 4];
        dst[0] = v.x; dst[1] = v.y; dst[2] = v.z; dst[3] = v.w;
    }
    for (int i = tid; i < BATCH * OT; i += THREADS) accs[i] = 0.0f;

    v8f acc00 = {}, acc01 = {};   // batch rows 0..15  x cols [0..15],[16..31]
    v8f acc10 = {}, acc11 = {};   // batch rows 16..31 x cols [0..15],[16..31]

    const int n     = lane & 15;              // column within half-tile
    const int khalf = (lane >> 4) << 1;       // 0 (lanes 0-15) or 2 (lanes 16-31)
    const float* xrow_lo = &xs[n * XS_STRIDE];
    const float* xrow_hi = &xs[(n + 16) * XS_STRIDE];

    // ---- per-lane incremental triu index state (k stride = QSTEP) ----
    // lane's k sequence: kc0 + wave*4 + khalf + QSTEP*i  (always even)
    int q = kc0 + wave * 4 + khalf - IN_DIM;  // current step's q (may be <0 in chunk 0)
    int r, c;
    {
        int m  = (q < 0) ? ((-q) + QSTEP - 1) / QSTEP : 0;
        int qi = q + m * QSTEP;               // first non-negative q in sequence
        triu_solve(qi, r, c);
    }

    const int wrow = tid >> 4;                // 0..31 : W row within tile
    const int wseg = (tid & 15) * 16;         // 16-float k-segment within KT

    for (int t = 0; t < ntiles; ++t) {
        const int kt0 = kc0 + t * KT;
        __syncthreads();
        // ---- coalesced stage: W[o0+wrow][kt0+wseg..+15] -> wtile[wrow][wseg..] ----
        {
            const float* src = W + (size_t)(o0 + wrow) * K_TOT + (size_t)kt0 + wseg;
            float* dst = &wtile[wrow * WSTRIDE + wseg];
#pragma unroll
            for (int u4 = 0; u4 < 4; ++u4)
                ((float4*)dst)[u4] = ((const float4*)src)[u4];   // ds_store_b128
            if (t + 1 < ntiles)
                __builtin_prefetch(src + KT, 0, 1);              // global_prefetch_b8
        }
        __syncthreads();

        // ---- 4 k-steps of 4 per wave: D += A(16x4) x B(4x16), 4 WMMAs/step ----
#pragma unroll
        for (int j = 0; j < 4; ++j) {
            const int ktb = (wave << 2) + (j << 6) + khalf;  // lane's kt base

            // W^T fragments: contiguous pair -> ds_load_b64, conflict-free
            const float2 b0 = *(const float2*)&wtile[n * WSTRIDE + ktb];
            const float2 b1 = *(const float2*)&wtile[(n + 16) * WSTRIDE + ktb];
            v2f bf0, bf1;
            bf0[0] = b0.x; bf0[1] = b0.y;
            bf1[0] = b1.x; bf1[1] = b1.y;

            // feature fragments at q and q+1 (both linear or both quadratic)
            float f0lo, f0hi, f1lo, f1hi;
            if (q < 0) {
                const int kg = IN_DIM + q;
                f0lo = xrow_lo[kg];     f0hi = xrow_hi[kg];
                f1lo = xrow_lo[kg + 1]; f1hi = xrow_hi[kg + 1];
            } else {
                const int r2 = (c < IN_DIM - 1) ? r : r + 1;
                const int c2 = (c < IN_DIM - 1) ? c + 1 : r + 1;
                f0lo = xrow_lo[r]  * xrow_lo[c];
                f0hi = xrow_hi[r]  * xrow_hi[c];
                f1lo = xrow_lo[r2] * xrow_lo[c2];
                f1hi = xrow_hi[r2] * xrow_hi[c2];
                // advance (r,c) by QSTEP along the triu sequence
                c += QSTEP;
                while (c >= IN_DIM && r < IN_DIM - 1) { ++r; c += r - IN_DIM; }
            }
            q += QSTEP;

            v2f a0, a1;
            a0[0] = f0lo; a0[1] = f1lo;   // batch rows 0..15
            a1[0] = f0hi; a1[1] = f1hi;   // batch rows 16..31

            acc00 = __builtin_amdgcn_wmma_f32_16x16x4_f32(
                        false, a0, false, bf0, (short)0, acc00, false, false);
            acc01 = __builtin_amdgcn_wmma_f32_16x16x4_f32(
                        false, a0, false, bf1, (short)0, acc01, false, false);
            acc10 = __builtin_amdgcn_wmma_f32_16x16x4_f32(
                        false, a1, false, bf0, (short)0, acc10, false, false);
            acc11 = __builtin_amdgcn_wmma_f32_16x16x4_f32(
                        false, a1, false, bf1, (short)0, acc11, false, false);
        }
    }

    // ---- cross-wave reduction in LDS (ds_add_f32), then global atomics ----
    __syncthreads();
    const int mrow = (lane < 16) ? 0 : 8;
#pragma unroll
    for (int v = 0; v < 8; ++v) {
        atomicAdd(&accs[(v + mrow) * OT + n],           acc00[v]);
        atomicAdd(&accs[(v + mrow) * OT + n + 16],      acc01[v]);
        atomicAdd(&accs[(v + mrow + 16) * OT + n],      acc10[v]);
        atomicAdd(&accs[(v + mrow + 16) * OT + n + 16], acc11[v]);
    }
    __syncthreads();
    for (int i = tid; i < BATCH * OT; i += THREADS) {
        int b  = i >> 5;        // /OT
        int nn = i & (OT - 1);
        atomicAdd(&out[b * OUT_DIM + o0 + nn], accs[i]);
    }
}

extern "C" void kernel_launch(void* const* d_in, const int* in_sizes, int n_in,
                              void* d_out, int out_size, void* d_ws, size_t ws_size,
                              hipStream_t stream) {
    const float* X    = (const float*)d_in[0];   // [32,64,16] == [32,1024]
    const float* W    = (const float*)d_in[1];   // [512, 525824]
    const float* bias = (const float*)d_in[2];   // [512]
    float* out = (float*)d_out;                  // [32,512]

    bias_init_kernel<<<(BATCH * OUT_DIM + 255) / 256, 256, 0, stream>>>(bias, out);

    dim3 grid(OUT_DIM / OT, NKC);                // 16 x 65 workgroups
    quad_gemm_kernel<<<grid, THREADS, 0, stream>>>(X, W, out);
}